// LossGIoU_7413113553025
// MI455X (gfx1250) — compile-verified
//
#include <hip/hip_runtime.h>

typedef __attribute__((ext_vector_type(2))) float v2f;
typedef __attribute__((ext_vector_type(8))) float v8f;

struct Accum {
    float loss;
    float pos;
    unsigned int done;
};

// Exact f32 wave32 sum using V_WMMA_F32_16X16X4_F32.
// A (16x4, f32): lane L supplies row (L&15); VGPR a[0] -> K = (L<16 ? 0 : 2),
//                a[1] -> K = (L<16 ? 1 : 3).  We place val in a[0], 0 in a[1].
// B (4x16) = all ones (layout independent: every slot = 1.0f).
// => D[m,n] = val[m] + val[m+16] for every n.
// C/D layout: c[j] lanes 0-15 hold M=j, lanes 16-31 hold M=j+8.
// sum_j c[j] gives (lanes 0-15): sum_{m=0..7} (val[m]+val[m+16])
//                  (lanes 16-31): sum_{m=8..15}(val[m]+val[m+16])
// One xor-16 shuffle-add completes the full 32-lane sum in every lane.
__device__ __forceinline__ float wave_sum_f32_wmma(float val) {
    v2f a;
    a[0] = val;
    a[1] = 0.0f;
    v2f b;
    b[0] = 1.0f;
    b[1] = 1.0f;
    v8f c = {};
    c = __builtin_amdgcn_wmma_f32_16x16x4_f32(
        /*neg_a=*/false, a, /*neg_b=*/false, b,
        /*c_mod=*/(short)0, c, /*reuse_a=*/false, /*reuse_b=*/false);
    float s = ((c[0] + c[1]) + (c[2] + c[3])) + ((c[4] + c[5]) + (c[6] + c[7]));
    s += __shfl_xor(s, 16, 32);
    return s;
}

__global__ void giou_init_ws(Accum* ws) {
    ws->loss = 0.0f;
    ws->pos = 0.0f;
    ws->done = 0u;
}

__global__ void giou_loss_kernel(const float4* __restrict__ pd,   // [B*N] deltas
                                 const float4* __restrict__ tb,   // [B*N] target boxes
                                 const float4* __restrict__ an,   // [N] anchors
                                 const unsigned char* __restrict__ mask,  // [B*N] bool
                                 float* __restrict__ out,
                                 Accum* __restrict__ ws,
                                 int total, int N) {
    const int tid = blockIdx.x * blockDim.x + threadIdx.x;
    const int stride = gridDim.x * blockDim.x;

    float lossAcc = 0.0f;
    float posAcc = 0.0f;

    int n = tid % N;  // anchor index for first element; tracked incrementally
    for (int i = tid; i < total; i += stride) {
        const float4 d = pd[i];      // (dx, dy, dw, dh)
        const float4 t = tb[i];      // (x1, y1, x2, y2)
        const float4 a = an[n];      // (x1, y1, x2, y2)
        const float m = mask[i] ? 1.0f : 0.0f;

        // delta2bbox
        const float w  = a.z - a.x;
        const float h  = a.w - a.y;
        const float cx = a.x + 0.5f * w;
        const float cy = a.y + 0.5f * h;
        const float pcx = d.x * w + cx;
        const float pcy = d.y * h + cy;
        const float pw  = __expf(d.z) * w;
        const float ph  = __expf(d.w) * h;
        const float px1 = pcx - 0.5f * pw;
        const float py1 = pcy - 0.5f * ph;
        const float px2 = pcx + 0.5f * pw;
        const float py2 = pcy + 0.5f * ph;

        // elementwise GIoU
        const float area1 = (px2 - px1) * (py2 - py1);
        const float area2 = (t.z - t.x) * (t.w - t.y);
        const float ltx = fmaxf(px1, t.x);
        const float lty = fmaxf(py1, t.y);
        const float rbx = fminf(px2, t.z);
        const float rby = fminf(py2, t.w);
        const float iw = fmaxf(rbx - ltx, 0.0f);
        const float ih = fmaxf(rby - lty, 0.0f);
        const float inter = iw * ih;
        const float uni = area1 + area2 - inter;
        const float iou = inter / uni;
        const float ex1 = fminf(px1, t.x);
        const float ey1 = fminf(py1, t.y);
        const float ex2 = fmaxf(px2, t.z);
        const float ey2 = fmaxf(py2, t.w);
        const float ew = fmaxf(ex2 - ex1, 0.0f);
        const float eh = fmaxf(ey2 - ey1, 0.0f);
        const float enc = ew * eh;
        const float giou = iou - (enc - uni) / enc;

        lossAcc += (1.0f - giou) * m;
        posAcc += m;

        // advance anchor index: (n + stride) mod N without integer division
        n += stride;
        while (n >= N) n -= N;
    }

    // Wave32 reduction via V_WMMA_F32_16X16X4_F32 (exact f32).
    const float wl = wave_sum_f32_wmma(lossAcc);
    const float wp = wave_sum_f32_wmma(posAcc);

    if ((threadIdx.x & 31) == 0) {
        atomicAdd(&ws->loss, wl);
        atomicAdd(&ws->pos, wp);
    }
    __syncthreads();

    if (threadIdx.x == 0) {
        __threadfence();
        const unsigned int prev = atomicAdd(&ws->done, 1u);
        if (prev == gridDim.x - 1) {
            // Coherent read-back of the fully accumulated totals.
            const float L = atomicAdd(&ws->loss, 0.0f);
            const float P = atomicAdd(&ws->pos, 0.0f);
            out[0] = L / fmaxf(P, 1.0f);
        }
    }
}

extern "C" void kernel_launch(void* const* d_in, const int* in_sizes, int n_in,
                              void* d_out, int out_size, void* d_ws, size_t ws_size,
                              hipStream_t stream) {
    const float4* pd = (const float4*)d_in[0];         // pred_deltas  [B,N,4] f32
    const float4* tb = (const float4*)d_in[1];         // target_boxes [B,N,4] f32
    const float4* an = (const float4*)d_in[2];         // anchors      [N,4]   f32
    const unsigned char* mask = (const unsigned char*)d_in[3];  // pos_mask [B,N] bool (1B)

    const int N = in_sizes[2] / 4;       // 262144
    const int total = in_sizes[3];       // B*N = 4,194,304
    Accum* ws = (Accum*)d_ws;

    giou_init_ws<<<1, 1, 0, stream>>>(ws);

    const int threads = 256;  // 8 waves per block on wave32
    int blocks = (total + threads * 16 - 1) / (threads * 16);  // ~16 elems/thread
    if (blocks > 2048) blocks = 2048;
    if (blocks < 1) blocks = 1;

    giou_loss_kernel<<<blocks, threads, 0, stream>>>(pd, tb, an, mask,
                                                     (float*)d_out, ws, total, N);
}